// REN_63909113364565
// MI455X (gfx1250) — compile-verified
//
#include <hip/hip_runtime.h>
#include <hip/hip_bf16.h>

typedef __attribute__((ext_vector_type(2))) float v2f;
typedef __attribute__((ext_vector_type(8))) float v8f;

#define DIMX 256
#define LDIM 256
#define DIN  128
#define DOUT 128
#define BATCH 8192

// Branchless tanh on native transcendentals: v_exp_f32 + v_rcp_f32.
// tanh(x) = 1 - 2/(exp(2x)+1); saturates correctly for |x| large.
__device__ __forceinline__ float fast_tanh(float x) {
    const float e = __builtin_amdgcn_exp2f(x * 2.8853900817779268f); // 2*log2(e)
    return 1.0f - 2.0f * __builtin_amdgcn_rcpf(e + 1.0f);
}

// ---------------------------------------------------------------------------
// In-place Gauss-Jordan inverse of E (256x256) in LDS. E is strongly
// diagonally dominant (diag ~192, offdiag sigma ~7) so no pivoting needed.
// One workgroup, 1024 threads, 256KB dynamic LDS (320KB/WGP on CDNA5).
// ---------------------------------------------------------------------------
__global__ void inv_kernel(const float* __restrict__ E, float* __restrict__ Einv) {
    extern __shared__ float A[];           // 256*256 floats = 256 KB
    __shared__ float s_pinv;
    const int t = threadIdx.x;             // 0..1023
    for (int idx = t; idx < DIMX * DIMX; idx += 1024) A[idx] = E[idx];
    __syncthreads();
    for (int k = 0; k < DIMX; ++k) {
        if (t == 0) s_pinv = 1.0f / A[k * DIMX + k];
        __syncthreads();
        const float pinv = s_pinv;
        if (t < DIMX) {
            float v = (t == k) ? 1.0f : A[k * DIMX + t];
            A[k * DIMX + t] = v * pinv;    // scaled pivot row (identity threaded in-place)
        }
        __syncthreads();
        const int i  = t >> 2;             // 4 threads per row
        const int c0 = (t & 3) * 64;
        const float f = A[i * DIMX + k];   // pre-elimination multiplier
        __syncthreads();
        if (i != k) {
            #pragma unroll 8
            for (int c = c0; c < c0 + 64; ++c) {
                float v = (c == k) ? 0.0f : A[i * DIMX + c];
                A[i * DIMX + c] = v - f * A[k * DIMX + c];
            }
        }
        __syncthreads();
    }
    for (int idx = t; idx < DIMX * DIMX; idx += 1024) Einv[idx] = A[idx];
}

// ---------------------------------------------------------------------------
// cx = C1 @ x ; fx = Fm @ x ; rLam = 1/Lam       one block of 256 threads
// ---------------------------------------------------------------------------
__global__ void vec_kernel(const float* __restrict__ C1, const float* __restrict__ Fm,
                           const float* __restrict__ x,  const float* __restrict__ Lam,
                           float* __restrict__ cx, float* __restrict__ fx,
                           float* __restrict__ rLam) {
    const int l = threadIdx.x;
    float a = 0.0f, b = 0.0f;
    #pragma unroll 4
    for (int i = 0; i < DIMX; ++i) {
        const float xi = x[i];
        a = fmaf(C1[l * DIMX + i], xi, a);
        b = fmaf(Fm[l * DIMX + i], xi, b);
    }
    cx[l]   = a;
    fx[l]   = b;
    rLam[l] = 1.0f / Lam[l];               // exact divide once; scan multiplies
}

// ---------------------------------------------------------------------------
// Block-triangular REN scan with WMMA.
// W (256 x 256 batch cols) is LDS-resident (256 KB). For block p (16 rows):
//   V = D11[16p:16p+16, 0:16p] @ W[0:16p, :]      <- f32 WMMA, B from LDS
//   then a fully-unrolled 16-step diagonal micro-scan (thread-per-column,
//   branchless fast_tanh, reciprocal-Lam multiply).
// 8 waves, uniform control flow (EXEC all-ones at every WMMA).
// ---------------------------------------------------------------------------
__global__ void scan_kernel(const float* __restrict__ du,   // [L][BATCH]
                            const float* __restrict__ cx,   // [L]
                            const float* __restrict__ D11,  // [L][L] strict lower tri
                            const float* __restrict__ rLam, // [L] = 1/Lam
                            float* __restrict__ w_out) {    // [L][BATCH]
    extern __shared__ float lds[];
    float* Wl = lds;                       // 256 x 256 = 256 KB
    float* Vt = lds + 65536;               // 16 x 256  = 16 KB
    const int t    = threadIdx.x;          // 0..255
    const int lane = t & 31;
    const int wv   = t >> 5;               // 0..7
    const int half = lane >> 4;
    const int l15  = lane & 15;
    const int bCol = blockIdx.x * 256;

    #pragma unroll 1
    for (int p = 0; p < 16; ++p) {
        const int row0    = p * 16;
        const int colBase = wv * 32;       // each wave: 2 16-wide column tiles
        const int mA      = row0 + l15;
        v8f acc0 = {}; v8f acc1 = {};
        for (int k0 = 0; k0 < row0; k0 += 4) {
            const int ka = k0 + half * 2;
            v2f a, b0, b1;
            a.x  = D11[(size_t)mA * LDIM + ka];
            a.y  = D11[(size_t)mA * LDIM + ka + 1];
            b0.x = Wl[ka * 256 + colBase + l15];
            b0.y = Wl[(ka + 1) * 256 + colBase + l15];
            b1.x = Wl[ka * 256 + colBase + 16 + l15];
            b1.y = Wl[(ka + 1) * 256 + colBase + 16 + l15];
            acc0 = __builtin_amdgcn_wmma_f32_16x16x4_f32(false, a, false, b0,
                                                         (short)0, acc0, false, false);
            acc1 = __builtin_amdgcn_wmma_f32_16x16x4_f32(false, a, false, b1,
                                                         (short)0, acc1, false, false);
        }
        #pragma unroll
        for (int r = 0; r < 8; ++r) {      // dump V tile (16 x 256) to LDS
            const int m = r + half * 8;
            Vt[m * 256 + colBase + l15]      = acc0[r];
            Vt[m * 256 + colBase + 16 + l15] = acc1[r];
        }
        __syncthreads();
        {   // diagonal micro-scan: thread t owns batch column t
            const int b = bCol + t;
            float wloc[16];
            #pragma unroll
            for (int r = 0; r < 16; ++r) {
                const int l = row0 + r;
                float acc = cx[l] + du[(size_t)l * BATCH + b] + Vt[r * 256 + t];
                #pragma unroll
                for (int s = 0; s < r; ++s)           // D11 diag-block: s_load broadcast
                    acc = fmaf(D11[(size_t)l * LDIM + row0 + s], wloc[s], acc);
                wloc[r] = fast_tanh(acc * rLam[l]);
            }
            #pragma unroll
            for (int r = 0; r < 16; ++r) {
                Wl[(row0 + r) * 256 + t] = wloc[r];
                w_out[(size_t)(row0 + r) * BATCH + b] = wloc[r];
            }
        }
        __syncthreads();
    }
}

// ---------------------------------------------------------------------------
// f32 WMMA GEMM: C(MxN) = A(MxK,row-major) @ B + [bias over M] [+ C if ACC].
// Each wave computes a 16x64 strip: one A fragment feeds 4 independent
// accumulators per K-step (4 v_wmma in flight, 4x A reuse).
// TRANSB:  B element (k,n) at B[n*ldb+k]   STORE_T: C (m,n) at C[n*ldc+m]
// Block = 128 threads = 4 waves -> block tile 16 x 256.
// ---------------------------------------------------------------------------
template <bool TRANSB, bool ACC, bool STORE_T>
__global__ void gemm16(const float* __restrict__ A, int lda,
                       const float* __restrict__ B, int ldb,
                       float* __restrict__ C, int ldc,
                       const float* __restrict__ bias,
                       int M, int N, int K) {
    const int lane  = threadIdx.x & 31;
    const int wv    = threadIdx.x >> 5;
    const int half  = lane >> 4;
    const int l15   = lane & 15;
    const int nBase = blockIdx.x * 256 + wv * 64;
    const int mBase = blockIdx.y * 16;
    const int n0    = nBase + l15;
    const int mA    = mBase + l15;

    v8f acc[4];
    #pragma unroll
    for (int j = 0; j < 4; ++j) {
        const int n = n0 + j * 16;
        #pragma unroll
        for (int r = 0; r < 8; ++r) {
            const int m = mBase + r + half * 8;
            float v = 0.0f;
            if (ACC) v = STORE_T ? C[(size_t)n * ldc + m] : C[(size_t)m * ldc + n];
            if (bias) v += bias[m];
            acc[j][r] = v;
        }
    }

    for (int k0 = 0; k0 < K; k0 += 4) {
        const int ka = k0 + half * 2;
        v2f a;
        a.x = A[(size_t)mA * lda + ka];
        a.y = A[(size_t)mA * lda + ka + 1];
        #pragma unroll
        for (int j = 0; j < 4; ++j) {
            const int n = n0 + j * 16;
            v2f b;
            if (TRANSB) {
                b.x = B[(size_t)n * ldb + ka];
                b.y = B[(size_t)n * ldb + ka + 1];
            } else {
                b.x = B[(size_t)ka * ldb + n];
                b.y = B[(size_t)(ka + 1) * ldb + n];
                if ((k0 & 12) == 0 && k0 + 16 < K)   // gfx1250 global_prefetch_b8
                    __builtin_prefetch((const void*)&B[(size_t)(k0 + 16) * ldb + n], 0, 1);
            }
            acc[j] = __builtin_amdgcn_wmma_f32_16x16x4_f32(false, a, false, b,
                                                           (short)0, acc[j], false, false);
        }
    }

    #pragma unroll
    for (int j = 0; j < 4; ++j) {
        const int n = n0 + j * 16;
        #pragma unroll
        for (int r = 0; r < 8; ++r) {
            const int m = mBase + r + half * 8;
            if (STORE_T) C[(size_t)n * ldc + m] = acc[j][r];
            else         C[(size_t)m * ldc + n] = acc[j][r];
        }
    }
}

// ---------------------------------------------------------------------------
extern "C" void kernel_launch(void* const* d_in, const int* in_sizes, int n_in,
                              void* d_out, int out_size, void* d_ws, size_t ws_size,
                              hipStream_t stream) {
    (void)in_sizes; (void)n_in; (void)out_size; (void)ws_size;
    const float* u   = (const float*)d_in[0];   // (8192,1,128)
    const float* x   = (const float*)d_in[1];   // (1,1,256)
    const float* Fm  = (const float*)d_in[2];   // (256,256)
    const float* B1  = (const float*)d_in[3];   // (256,256)
    const float* B2  = (const float*)d_in[4];   // (256,128)
    const float* C1  = (const float*)d_in[5];   // (256,256)
    const float* C2  = (const float*)d_in[6];   // (128,256)
    const float* D11 = (const float*)d_in[7];   // (256,256)
    const float* D12 = (const float*)d_in[8];   // (256,128)
    const float* D21 = (const float*)d_in[9];   // (128,256)
    const float* D22 = (const float*)d_in[10];  // (128,128)
    const float* E   = (const float*)d_in[11];  // (256,256)
    const float* Lam = (const float*)d_in[12];  // (256,)
    float* out = (float*)d_out;                 // (8192,1,128)

    float* ws   = (float*)d_ws;
    float* T    = ws;                           // 256*8192  (du, later t^T)
    float* W    = ws + 2097152;                 // 256*8192
    float* Einv = ws + 2 * 2097152;             // 256*256
    float* M1   = Einv + 65536;                 // 128*256  (C2 @ Einv)
    float* cx   = M1 + 32768;                   // 256
    float* fx   = cx + 256;                     // 256
    float* rLam = fx + 256;                     // 256

    const size_t lds256k = (size_t)DIMX * DIMX * sizeof(float);       // 256 KB
    const size_t ldsScan = lds256k + 16 * 256 * sizeof(float);        // 272 KB

    // one-off small work
    inv_kernel<<<1, 1024, lds256k, stream>>>(E, Einv);
    vec_kernel<<<1, 256, 0, stream>>>(C1, Fm, x, Lam, cx, fx, rLam);

    // du = D12 @ U^T                      (256 x 8192, K=128)
    gemm16<true, false, false><<<dim3(BATCH / 256, LDIM / 16), 128, 0, stream>>>(
        D12, DIN, u, DIN, T, BATCH, nullptr, LDIM, BATCH, DIN);

    // w (block-triangular WMMA recurrence)
    scan_kernel<<<BATCH / 256, 256, ldsScan, stream>>>(T, cx, D11, rLam, W);

    // T = B1 @ W + fx                     (256 x 8192, K=256)
    gemm16<false, false, false><<<dim3(BATCH / 256, DIMX / 16), 128, 0, stream>>>(
        B1, LDIM, W, BATCH, T, BATCH, fx, DIMX, BATCH, LDIM);
    // T += B2 @ U^T                       (K=128)
    gemm16<true, true, false><<<dim3(BATCH / 256, DIMX / 16), 128, 0, stream>>>(
        B2, DIN, u, DIN, T, BATCH, nullptr, DIMX, BATCH, DIN);

    // M1 = C2 @ Einv                      (128 x 256, K=256) — fuses away x_new
    gemm16<false, false, false><<<dim3(DIMX / 256, DOUT / 16), 128, 0, stream>>>(
        C2, DIMX, Einv, DIMX, M1, DIMX, nullptr, DOUT, DIMX, DIMX);

    // y^T = M1 @ T ; += D21 @ W ; += D22 @ U^T   (stored transposed -> (b, o))
    gemm16<false, false, true><<<dim3(BATCH / 256, DOUT / 16), 128, 0, stream>>>(
        M1, DIMX, T, BATCH, out, DOUT, nullptr, DOUT, BATCH, DIMX);
    gemm16<false, true, true><<<dim3(BATCH / 256, DOUT / 16), 128, 0, stream>>>(
        D21, LDIM, W, BATCH, out, DOUT, nullptr, DOUT, BATCH, LDIM);
    gemm16<true, true, true><<<dim3(BATCH / 256, DOUT / 16), 128, 0, stream>>>(
        D22, DIN, u, DIN, out, DOUT, nullptr, DOUT, BATCH, DIN);
}